// BaselineClassifier_3092376453140
// MI455X (gfx1250) — compile-verified
//
#include <hip/hip_runtime.h>
#include <hip/hip_bf16.h>

#define N_NODES  100000
#define N_EDGES  1600000
#define IN_DIM   128
#define HID      64
#define OUT_DIM  16
#define BN_EPS   1e-5f

typedef __attribute__((ext_vector_type(2))) float v2f;
typedef __attribute__((ext_vector_type(8))) float v8f;

static __device__ __forceinline__ void atomic_add_f32(float* p, float v) {
    (void)__hip_atomic_fetch_add(p, v, __ATOMIC_RELAXED, __HIP_MEMORY_SCOPE_AGENT);
}

// -------- generic zero fill (grid-stride) --------
__global__ void zero_kernel(float* __restrict__ p, unsigned long long n) {
    unsigned long long i = (unsigned long long)blockIdx.x * blockDim.x + threadIdx.x;
    unsigned long long stride = (unsigned long long)gridDim.x * blockDim.x;
    for (; i < n; i += stride) p[i] = 0.0f;
}

// -------- degree: deg[dst[e]] += 1 --------
__global__ void degree_kernel(const int* __restrict__ dst, float* __restrict__ deg) {
    int e = blockIdx.x * blockDim.x + threadIdx.x;
    if (e < N_EDGES) atomic_add_f32(&deg[dst[e]], 1.0f);
}

// -------- deg -> 1/max(deg,1) in place --------
__global__ void deginv_kernel(float* __restrict__ deg) {
    int i = blockIdx.x * blockDim.x + threadIdx.x;
    if (i < N_NODES) deg[i] = 1.0f / fmaxf(deg[i], 1.0f);
}

// -------- GEMM1: y1 = x @ W1l^T, xr = x @ W1r^T  (f32 WMMA 16x16x4) --------
// Block = 128 threads = 4 waves; wave w owns output columns [16w,16w+16) of both mats.
// A (16x4 f32): lane<16 -> M=lane, VGPR0=K0,VGPR1=K1 ; lane>=16 -> M=lane-16, K2/K3.
// B (4x16 f32): mirror layout; B[k][n] = W[n_global*IN_DIM + k].
// C (16x16 f32): vgpr v -> row v + 8*(lane>=16), col = lane&15.
__global__ __launch_bounds__(128) void gemm1_wmma(
    const float* __restrict__ x,    // [N,128]
    const float* __restrict__ W1l,  // [64,128]
    const float* __restrict__ W1r,  // [64,128]
    float* __restrict__ y1,         // [N,64]
    float* __restrict__ xr)         // [N,64]
{
    const int tile = blockIdx.x;              // 16-node tile (N/16 = 6250 exact)
    const int wave = threadIdx.x >> 5;        // 0..3 -> feature tile
    const int lane = threadIdx.x & 31;
    const int half = lane >> 4;               // 0/1 selects K pair
    const int l16  = lane & 15;
    const int featbase = wave * 16;
    const int arow = tile * 16 + l16;

    const float* xrow = x   + (size_t)arow * IN_DIM + half * 2;
    const float* wl   = W1l + (size_t)(featbase + l16) * IN_DIM + half * 2;
    const float* wr   = W1r + (size_t)(featbase + l16) * IN_DIM + half * 2;

    v8f cl = {}; v8f cr = {};
    #pragma unroll
    for (int k = 0; k < IN_DIM; k += 4) {
        v2f a  = *(const v2f*)(xrow + k);
        v2f bl = *(const v2f*)(wl + k);
        v2f br = *(const v2f*)(wr + k);
        cl = __builtin_amdgcn_wmma_f32_16x16x4_f32(false, a, false, bl, (short)0, cl, false, false);
        cr = __builtin_amdgcn_wmma_f32_16x16x4_f32(false, a, false, br, (short)0, cr, false, false);
    }
    #pragma unroll
    for (int v = 0; v < 8; ++v) {
        int r = tile * 16 + v + 8 * half;
        size_t o = (size_t)r * HID + featbase + l16;
        y1[o] = cl[v];
        xr[o] = cr[v];
    }
}

// -------- edge aggregation: agg[dst[e]*F + f] += y[src[e]*F + f] --------
__global__ void agg_kernel(const int* __restrict__ src, const int* __restrict__ dst,
                           const float* __restrict__ y, float* __restrict__ agg,
                           int log2F) {
    long long idx = (long long)blockIdx.x * blockDim.x + threadIdx.x;
    long long total = (long long)N_EDGES << log2F;
    if (idx >= total) return;
    int e = (int)(idx >> log2F);
    int f = (int)(idx & ((1 << log2F) - 1));
    int s = src[e], d = dst[e];
    atomic_add_f32(&agg[((size_t)d << log2F) + f], y[((size_t)s << log2F) + f]);
}

// -------- h_pre = agg1*deginv + b1l + xr (in place into agg1) + BN stats --------
__global__ __launch_bounds__(256) void hpre_stats_kernel(
    float* __restrict__ agg1,          // [N,64] in: sums, out: h_pre
    const float* __restrict__ xr,      // [N,64]
    const float* __restrict__ deginv,  // [N]
    const float* __restrict__ b1l,     // [64]
    float* __restrict__ stats)         // [0:64]=sum, [64:128]=sumsq
{
    const int j = threadIdx.x & 63;
    const int r = threadIdx.x >> 6;    // 0..3 nodes per block-iteration
    const float bj = b1l[j];
    float s = 0.0f, q = 0.0f;
    for (int n = blockIdx.x * 4 + r; n < N_NODES; n += gridDim.x * 4) {
        size_t o = (size_t)n * HID + j;
        float v = agg1[o] * deginv[n] + bj + xr[o];
        agg1[o] = v;
        s += v; q += v * v;
    }
    __shared__ float ss[256], sq[256];
    ss[threadIdx.x] = s; sq[threadIdx.x] = q;
    __syncthreads();
    if (r == 0) {
        s = ss[j] + ss[j + 64] + ss[j + 128] + ss[j + 192];
        q = sq[j] + sq[j + 64] + sq[j + 128] + sq[j + 192];
        atomic_add_f32(&stats[j], s);
        atomic_add_f32(&stats[64 + j], q);
    }
}

// -------- fold BN into per-feature scale/shift --------
__global__ void bnparams_kernel(const float* __restrict__ stats,
                                const float* __restrict__ gamma,
                                const float* __restrict__ beta,
                                float* __restrict__ scsh) { // [0:64]=scale,[64:128]=shift
    int j = threadIdx.x;
    if (j >= HID) return;
    float invN = 1.0f / (float)N_NODES;
    float mean = stats[j] * invN;
    float var  = stats[64 + j] * invN - mean * mean;
    float rs   = rsqrtf(var + BN_EPS);
    float sc   = rs * gamma[j];
    scsh[j]      = sc;
    scsh[64 + j] = beta[j] - mean * sc;
}

// -------- GEMM2: h = relu(hpre*scale+shift) applied on the fly;
//          y2 = h @ W2l^T, hr = h @ W2r^T  (f32 WMMA 16x16x4) --------
__global__ __launch_bounds__(256) void gemm2_wmma(
    const float* __restrict__ hpre,   // [N,64]
    const float* __restrict__ scale,  // [64]
    const float* __restrict__ shift,  // [64]
    const float* __restrict__ W2l,    // [16,64]
    const float* __restrict__ W2r,    // [16,64]
    float* __restrict__ y2,           // [N,16]
    float* __restrict__ hr)           // [N,16]
{
    const int tile = blockIdx.x * 8 + (threadIdx.x >> 5);   // one 16-node tile per wave
    if (tile >= (N_NODES / 16)) return;                      // wave-uniform -> EXEC all-1s
    const int lane = threadIdx.x & 31;
    const int half = lane >> 4;
    const int l16  = lane & 15;
    const int arow = tile * 16 + l16;

    const float* hrow = hpre + (size_t)arow * HID + half * 2;
    const float* sc   = scale + half * 2;
    const float* sh   = shift + half * 2;
    const float* wl   = W2l + (size_t)l16 * HID + half * 2;
    const float* wr   = W2r + (size_t)l16 * HID + half * 2;

    v8f cl = {}; v8f cr = {};
    #pragma unroll
    for (int k = 0; k < HID; k += 4) {
        v2f a;
        a.x = fmaxf(hrow[k]     * sc[k]     + sh[k],     0.0f);
        a.y = fmaxf(hrow[k + 1] * sc[k + 1] + sh[k + 1], 0.0f);
        v2f bl = *(const v2f*)(wl + k);
        v2f br = *(const v2f*)(wr + k);
        cl = __builtin_amdgcn_wmma_f32_16x16x4_f32(false, a, false, bl, (short)0, cl, false, false);
        cr = __builtin_amdgcn_wmma_f32_16x16x4_f32(false, a, false, br, (short)0, cr, false, false);
    }
    #pragma unroll
    for (int v = 0; v < 8; ++v) {
        int r = tile * 16 + v + 8 * half;
        size_t o = (size_t)r * OUT_DIM + l16;
        y2[o] = cl[v];
        hr[o] = cr[v];
    }
}

// -------- final: emb = agg2*deginv + b2l + hr ; logits = emb @ Wc^T + bc --------
__global__ __launch_bounds__(256) void final_kernel(
    const float* __restrict__ agg2,    // [N,16]
    const float* __restrict__ deginv,  // [N]
    const float* __restrict__ b2l,     // [16]
    const float* __restrict__ Wc,      // [2,16]
    const float* __restrict__ bc,      // [2]
    float* __restrict__ out_logits,    // [N,2]
    float* __restrict__ out_emb)       // [N,16] (holds hr on entry)
{
    int n = blockIdx.x * blockDim.x + threadIdx.x;
    if (n >= N_NODES) return;
    float di = deginv[n];
    float l0 = bc[0], l1 = bc[1];
    float emb[OUT_DIM];
    #pragma unroll
    for (int j = 0; j < OUT_DIM; ++j) {
        float v = agg2[(size_t)n * OUT_DIM + j] * di + b2l[j] + out_emb[(size_t)n * OUT_DIM + j];
        emb[j] = v;
        l0 += v * Wc[j];
        l1 += v * Wc[OUT_DIM + j];
    }
    #pragma unroll
    for (int j = 0; j < OUT_DIM; ++j) out_emb[(size_t)n * OUT_DIM + j] = emb[j];
    out_logits[(size_t)n * 2]     = l0;
    out_logits[(size_t)n * 2 + 1] = l1;
}

extern "C" void kernel_launch(void* const* d_in, const int* in_sizes, int n_in,
                              void* d_out, int out_size, void* d_ws, size_t ws_size,
                              hipStream_t stream) {
    (void)in_sizes; (void)n_in; (void)out_size; (void)ws_size;
    const float* x     = (const float*)d_in[0];
    const int*   src   = (const int*)  d_in[1];
    const int*   dst   = (const int*)  d_in[2];
    const float* W1l   = (const float*)d_in[3];
    const float* b1l   = (const float*)d_in[4];
    const float* W1r   = (const float*)d_in[5];
    const float* gamma = (const float*)d_in[6];
    const float* beta  = (const float*)d_in[7];
    const float* W2l   = (const float*)d_in[8];
    const float* b2l   = (const float*)d_in[9];
    const float* W2r   = (const float*)d_in[10];
    const float* Wc    = (const float*)d_in[11];
    const float* bc    = (const float*)d_in[12];

    // Workspace layout (floats): deg[N] | y1[N*64] | xr[N*64] | agg1[N*64] | stats[256]
    float* ws    = (float*)d_ws;
    float* deg   = ws;
    float* y1    = deg  + N_NODES;
    float* xr    = y1   + (size_t)N_NODES * HID;
    float* agg1  = xr   + (size_t)N_NODES * HID;
    float* stats = agg1 + (size_t)N_NODES * HID;   // 256 floats: sum|sumsq|scale|shift
    float* y2    = y1;                              // reuse (y1 dead after agg1)
    float* agg2  = xr;                              // reuse (xr dead after hpre_stats)

    float* out_logits = (float*)d_out;
    float* out_emb    = out_logits + (size_t)N_NODES * 2;  // hr staged here, then emb

    zero_kernel<<<1024, 256, 0, stream>>>(deg, N_NODES);
    zero_kernel<<<2048, 256, 0, stream>>>(agg1, (unsigned long long)N_NODES * HID);
    zero_kernel<<<1, 256, 0, stream>>>(stats, 256);

    degree_kernel<<<(N_EDGES + 255) / 256, 256, 0, stream>>>(dst, deg);
    deginv_kernel<<<(N_NODES + 255) / 256, 256, 0, stream>>>(deg);

    gemm1_wmma<<<N_NODES / 16, 128, 0, stream>>>(x, W1l, W1r, y1, xr);

    {
        long long total = (long long)N_EDGES * HID;
        agg_kernel<<<(unsigned)((total + 255) / 256), 256, 0, stream>>>(src, dst, y1, agg1, 6);
    }

    hpre_stats_kernel<<<512, 256, 0, stream>>>(agg1, xr, deg, b1l, stats);
    bnparams_kernel<<<1, 64, 0, stream>>>(stats, gamma, beta, stats + 128);

    zero_kernel<<<256, 256, 0, stream>>>(agg2, (unsigned long long)N_NODES * OUT_DIM);

    gemm2_wmma<<<(N_NODES / 16 + 7) / 8, 256, 0, stream>>>(
        agg1, stats + 128, stats + 192, W2l, W2r, y2, out_emb);

    {
        long long total = (long long)N_EDGES * OUT_DIM;
        agg_kernel<<<(unsigned)((total + 255) / 256), 256, 0, stream>>>(src, dst, y2, agg2, 4);
    }

    final_kernel<<<(N_NODES + 255) / 256, 256, 0, stream>>>(
        agg2, deg, b2l, Wc, bc, out_logits, out_emb);
}